// SOS2Wavefront_39522289058145
// MI455X (gfx1250) — compile-verified
//
#include <hip/hip_runtime.h>
#include <math.h>

// Problem constants (match the reference).
namespace {
constexpr int   kNPts   = 256;
constexpr int   kNTheta = 180;
constexpr int   kNInt   = 500;
constexpr int   kGrid   = 512;
constexpr float kFov    = 0.06f;
constexpr float kRBody  = 0.05f;
constexpr float kV0     = 1500.0f;
constexpr int   kWavesPerBlock = 8;            // 256 threads / wave32
constexpr int   kPairs  = kNPts * kNTheta;     // 46080
}

// One wave32 per (point, theta) pair. Lanes stride over the 500 integration
// samples (consecutive lanes = adjacent chord samples -> near-coalesced
// gathers); per-wave ds-shuffle reduction; lane 0 writes wf[n,t].
__global__ __launch_bounds__(256) void sos2wavefront_kernel(
    const float* __restrict__ X,
    const float* __restrict__ Y,
    const float* __restrict__ SOS,
    float* __restrict__ out)
{
    const float kThetaStep = 6.283185307179586f / 179.0f;   // linspace(0,2pi,180)

    // thetas output: first 180 floats of d_out.
    if (blockIdx.x == 0 && threadIdx.x < kNTheta) {
        out[threadIdx.x] = (float)threadIdx.x * kThetaStep;
    }

    const int lane = threadIdx.x & 31;
    const int wave = threadIdx.x >> 5;
    const int pair = blockIdx.x * kWavesPerBlock + wave;
    if (pair >= kPairs) return;
    const int n = pair / kNTheta;
    const int t = pair - n * kNTheta;

    const float xv = X[n];
    const float yv = Y[n];
    const float theta = (float)t * kThetaStep;

    // ---- chord geometry (uniform across the wave, computed once) ----
    const float r   = sqrtf(xv * xv + yv * yv);
    const float phi = atan2f(xv, yv);            // NOTE: reference uses arctan2(x, y)
    const float a   = theta - phi;
    const float s   = r * sinf(a);
    const float c   = cosf(a);                    // cos even: cos(phi-theta)==cos(a)
    const float disc = kRBody * kRBody - s * s;
    const float sq   = sqrtf(fmaxf(disc, 0.0f));
    const bool inside = r < kRBody;
    const bool hit    = (c >= 0.0f) && (kRBody >= fabsf(s));
    const float l = inside ? (sq + r * c) : (hit ? (sq + r * c) : 0.0f);

    const float sth = sinf(theta);
    const float cth = cosf(theta);

    const float invK  = 1.0f / (float)(kNInt - 1);
    const float dstep = l * invK;                         // uniform path spacing
    const float dxg   = 2.0f * kFov / (float)(kGrid - 1);
    const float invDx = 1.0f / dxg;

    // Folded per-wave index coefficients:
    //   jx = rint(cx - ax*k),  iy = rint(cy + ay*k)
    // (rint(-v) == -rint(v) exactly under round-half-even, matching jnp.round)
    const float cx = (xv + kFov) * invDx;
    const float cy = (kFov - yv) * invDx;
    const float ax = dstep * sth * invDx;
    const float ay = dstep * cth * invDx;

    // ---- line integral: weighted gather-sum over K samples ----
    float acc = 0.0f;
    for (int k0 = lane; k0 < kNInt; k0 += 128) {
        // One CDNA5 global_prefetch_b8 per 4 gathers, 128 samples (~4
        // iterations) of lead time; indices clamped so always in-bounds.
        {
            const float kp = (float)(k0 + 128);
            const int jp = min(max((int)rintf(cx - ax * kp), 0), kGrid - 1);
            const int ip = min(max((int)rintf(cy + ay * kp), 0), kGrid - 1);
            __builtin_prefetch(SOS + (ip * kGrid + jp), 0, 3);
        }
        #pragma unroll
        for (int u = 0; u < 4; ++u) {
            const int k = k0 + 32 * u;
            if (k < kNInt) {
                const float kf = (float)k;
                int jx = (int)rintf(cx - ax * kf);         // v_rndne
                int iy = (int)rintf(cy + ay * kf);
                jx = min(max(jx, 0), kGrid - 1);            // v_med3
                iy = min(max(iy, 0), kGrid - 1);
                const float sv = SOS[iy * kGrid + jx];
                const float integ = 1.0f - __fdividef(kV0, sv);  // v_rcp + v_mul
                const float w = (k == 0 || k == kNInt - 1) ? 0.5f : 1.0f;
                acc = fmaf(w, integ, acc);
            }
        }
    }

    // ---- wave32 butterfly reduction (ds lane ops) ----
    #pragma unroll
    for (int off = 16; off > 0; off >>= 1)
        acc += __shfl_xor(acc, off, 32);

    if (lane == 0)
        out[kNTheta + pair] = acc * dstep;   // trapz with uniform spacing l/(K-1)
}

extern "C" void kernel_launch(void* const* d_in, const int* in_sizes, int n_in,
                              void* d_out, int out_size, void* d_ws, size_t ws_size,
                              hipStream_t stream) {
    (void)in_sizes; (void)n_in; (void)out_size; (void)d_ws; (void)ws_size;
    const float* x   = (const float*)d_in[0];
    const float* y   = (const float*)d_in[1];
    const float* sos = (const float*)d_in[2];
    float* out = (float*)d_out;

    const int blocks = kPairs / kWavesPerBlock;   // 46080 / 8 = 5760
    sos2wavefront_kernel<<<blocks, 256, 0, stream>>>(x, y, sos, out);
}